// DAMASK_GCN_14078902796913
// MI455X (gfx1250) — compile-verified
//
#include <hip/hip_runtime.h>

typedef __attribute__((ext_vector_type(2))) float v2f;
typedef __attribute__((ext_vector_type(4))) float v4f;
typedef __attribute__((ext_vector_type(8))) float v8f;

#define N_USERS 4096
#define N_ITEMS 8192
#define DIM     64
#define TILES_M (N_USERS / 16)  // 256
#define TILES_N (N_ITEMS / 16)  // 512
#define WAVES_PER_BLOCK 8
#define LOG2E 1.44269504088896340736f

// ---------------------------------------------------------------------------
// Prelude: dst[mat*n + row] = exp(emb_mat[row] . hvec)   (mat = blockIdx.y)
// ---------------------------------------------------------------------------
__global__ __launch_bounds__(256) void DAMASK_GCN_exp_dot_kernel(
    const float* __restrict__ e1, const float* __restrict__ e2,
    const float* __restrict__ e3, const float* __restrict__ hvec,
    float* __restrict__ dst, int n)
{
    const float* src = (blockIdx.y == 0) ? e1 : (blockIdx.y == 1) ? e2 : e3;
    const int row = blockIdx.x * blockDim.x + threadIdx.x;   // grid covers n exactly
    const float* p = src + (size_t)row * DIM;

    float acc = 0.f;
    #pragma unroll
    for (int j = 0; j < DIM / 4; ++j) {
        v4f a = *(const v4f*)(p + 4 * j);
        v4f hh = *(const v4f*)(hvec + 4 * j);
        acc = fmaf(a.x, hh.x, fmaf(a.y, hh.y, fmaf(a.z, hh.z, fmaf(a.w, hh.w, acc))));
    }
    dst[(size_t)blockIdx.y * n + row] = __builtin_amdgcn_exp2f(acc * LOG2E);
}

// ---------------------------------------------------------------------------
// Main: one wave32 per 16x16 output tile; three fp32 GEMM accumulators via
// chained V_WMMA_F32_16X16X4_F32; factorized-softmax + sigmoid epilogue.
//
// Virtual-K remap: lanes 0-15 carry actual k in [0,32), lanes 16-31 carry
// k in [32,64). WMMA slab 2j uses {4j,4j+1} / {32+4j,32+4j+1}; slab 2j+1 the
// .zw halves. Bijective over k=0..63, identical for A and B => exact dot.
// ---------------------------------------------------------------------------
__global__ __launch_bounds__(256) void DAMASK_GCN_att_rating_kernel(
    const float* __restrict__ u1, const float* __restrict__ i1,
    const float* __restrict__ u2, const float* __restrict__ i2,
    const float* __restrict__ u3, const float* __restrict__ i3,
    const float* __restrict__ Eu, const float* __restrict__ Ei,
    float* __restrict__ out)
{
    const int lane = threadIdx.x & 31;
    const int wave = threadIdx.x >> 5;
    const int tile = blockIdx.x * WAVES_PER_BLOCK + wave;
    const int tm   = tile / TILES_N;
    const int tn   = tile % TILES_N;
    const int hi   = lane >> 4;       // lane half
    const int l16  = lane & 15;

    const size_t koff = (size_t)(32 * hi);
    const float* pa1 = u1 + (size_t)(tm * 16 + l16) * DIM + koff;
    const float* pa2 = u2 + (size_t)(tm * 16 + l16) * DIM + koff;
    const float* pa3 = u3 + (size_t)(tm * 16 + l16) * DIM + koff;
    const float* pb1 = i1 + (size_t)(tn * 16 + l16) * DIM + koff;
    const float* pb2 = i2 + (size_t)(tn * 16 + l16) * DIM + koff;
    const float* pb3 = i3 + (size_t)(tn * 16 + l16) * DIM + koff;

    v8f acc1 = {}, acc2 = {}, acc3 = {};

    #pragma unroll
    for (int j = 0; j < 8; ++j) {
        v4f a1 = *(const v4f*)(pa1 + 4 * j);
        v4f b1 = *(const v4f*)(pb1 + 4 * j);
        v4f a2 = *(const v4f*)(pa2 + 4 * j);
        v4f b2 = *(const v4f*)(pb2 + 4 * j);
        v4f a3 = *(const v4f*)(pa3 + 4 * j);
        v4f b3 = *(const v4f*)(pb3 + 4 * j);

        acc1 = __builtin_amdgcn_wmma_f32_16x16x4_f32(false, __builtin_shufflevector(a1, a1, 0, 1),
                                                     false, __builtin_shufflevector(b1, b1, 0, 1),
                                                     (short)0, acc1, false, false);
        acc2 = __builtin_amdgcn_wmma_f32_16x16x4_f32(false, __builtin_shufflevector(a2, a2, 0, 1),
                                                     false, __builtin_shufflevector(b2, b2, 0, 1),
                                                     (short)0, acc2, false, false);
        acc3 = __builtin_amdgcn_wmma_f32_16x16x4_f32(false, __builtin_shufflevector(a3, a3, 0, 1),
                                                     false, __builtin_shufflevector(b3, b3, 0, 1),
                                                     (short)0, acc3, false, false);
        acc1 = __builtin_amdgcn_wmma_f32_16x16x4_f32(false, __builtin_shufflevector(a1, a1, 2, 3),
                                                     false, __builtin_shufflevector(b1, b1, 2, 3),
                                                     (short)0, acc1, false, false);
        acc2 = __builtin_amdgcn_wmma_f32_16x16x4_f32(false, __builtin_shufflevector(a2, a2, 2, 3),
                                                     false, __builtin_shufflevector(b2, b2, 2, 3),
                                                     (short)0, acc2, false, false);
        acc3 = __builtin_amdgcn_wmma_f32_16x16x4_f32(false, __builtin_shufflevector(a3, a3, 2, 3),
                                                     false, __builtin_shufflevector(b3, b3, 2, 3),
                                                     (short)0, acc3, false, false);
    }

    // Per-lane softmax factors. C/D layout: VGPR r -> (M = r + 8*hi, N = l16).
    // Each lane needs Eu[tm*16 + 8*hi .. +7] (8 contiguous) and Ei at its column.
    const float* Eu1 = Eu;
    const float* Eu2 = Eu + N_USERS;
    const float* Eu3 = Eu + 2 * N_USERS;
    const float* Ei1 = Ei;
    const float* Ei2 = Ei + N_ITEMS;
    const float* Ei3 = Ei + 2 * N_ITEMS;

    const int mbase = tm * 16 + 8 * hi;
    v4f eu1a = *(const v4f*)(Eu1 + mbase), eu1b = *(const v4f*)(Eu1 + mbase + 4);
    v4f eu2a = *(const v4f*)(Eu2 + mbase), eu2b = *(const v4f*)(Eu2 + mbase + 4);
    v4f eu3a = *(const v4f*)(Eu3 + mbase), eu3b = *(const v4f*)(Eu3 + mbase + 4);
    const float ei1 = Ei1[tn * 16 + l16];
    const float ei2 = Ei2[tn * 16 + l16];
    const float ei3 = Ei3[tn * 16 + l16];

    float* ocol = out + (size_t)(tm * 16 + 8 * hi) * N_ITEMS + tn * 16 + l16;

    #pragma unroll
    for (int r = 0; r < 8; ++r) {
        const float e1 = ((r < 4) ? eu1a[r & 3] : eu1b[r & 3]) * ei1;
        const float e2 = ((r < 4) ? eu2a[r & 3] : eu2b[r & 3]) * ei2;
        const float e3 = ((r < 4) ? eu3a[r & 3] : eu3b[r & 3]) * ei3;

        const float num = fmaf(acc1[r], e1, fmaf(acc2[r], e2, acc3[r] * e3));
        const float den = e1 + e2 + e3;
        const float x   = num * __builtin_amdgcn_rcpf(den);
        const float t   = __builtin_amdgcn_exp2f(-LOG2E * x);
        const float rating = __builtin_amdgcn_rcpf(1.0f + t);

        __builtin_nontemporal_store(rating, ocol + (size_t)r * N_ITEMS);
    }
}

extern "C" void kernel_launch(void* const* d_in, const int* in_sizes, int n_in,
                              void* d_out, int out_size, void* d_ws, size_t ws_size,
                              hipStream_t stream) {
    (void)in_sizes; (void)n_in; (void)out_size; (void)ws_size;
    const float* u1 = (const float*)d_in[0];
    const float* i1 = (const float*)d_in[1];
    const float* u2 = (const float*)d_in[2];
    const float* i2 = (const float*)d_in[3];
    const float* u3 = (const float*)d_in[4];
    const float* i3 = (const float*)d_in[5];
    const float* h  = (const float*)d_in[6];
    float* out = (float*)d_out;

    float* Eu = (float*)d_ws;               // 3 * 4096 floats
    float* Ei = Eu + 3 * N_USERS;           // 3 * 8192 floats   (total 147 KB)

    // Prelude: Eu_p = exp(u_p . h[:64]),  Ei_p = exp(i_p . h[64:])
    DAMASK_GCN_exp_dot_kernel<<<dim3(N_USERS / 256, 3), 256, 0, stream>>>(
        u1, u2, u3, h, Eu, N_USERS);
    DAMASK_GCN_exp_dot_kernel<<<dim3(N_ITEMS / 256, 3), 256, 0, stream>>>(
        i1, i2, i3, h + DIM, Ei, N_ITEMS);

    const int total_tiles = TILES_M * TILES_N;          // 131072
    const int blocks = total_tiles / WAVES_PER_BLOCK;   // 16384
    DAMASK_GCN_att_rating_kernel<<<blocks, 32 * WAVES_PER_BLOCK, 0, stream>>>(
        u1, i1, u2, i2, u3, i3, Eu, Ei, out);
}